// AdaConv_61246233640978
// MI455X (gfx1250) — compile-verified
//
#include <hip/hip_runtime.h>

typedef __attribute__((ext_vector_type(2))) float v2f;
typedef __attribute__((ext_vector_type(4))) float v4f;
typedef __attribute__((ext_vector_type(8))) float v8f;

#define B_DIM 32
#define C_DIM 256
#define HW_DIM 4096   // 64*64

// ---------------------------------------------------------------------------
// Fused 2-layer latent MLP for both paths (scale path: w1/b1/w2/b2,
// bias path: bw1/bb1/bw2/bb2) using V_WMMA_F32_16X16X4_F32.
//   hw      = relu(l @ W1^T + B1)        (kept in LDS, 2 x 32 x 256 f32)
//   result  = hw @ W2^T + B2             -> wsOut[path][b][o]
// One workgroup: 32 waves (wave32), wave w owns output tile
// (mtile = w/16 in 0..1, ntile = w%16 in 0..15), K-loop of 64 WMMAs.
// ---------------------------------------------------------------------------
__global__ __launch_bounds__(1024) void adaconv_mlp_kernel(
    const float* __restrict__ latent,
    const float* __restrict__ w1,  const float* __restrict__ b1,
    const float* __restrict__ w2,  const float* __restrict__ b2,
    const float* __restrict__ bw1, const float* __restrict__ bb1,
    const float* __restrict__ bw2, const float* __restrict__ bb2,
    float* __restrict__ wsOut)     // [2][32][256]: slot 0 = scale, 1 = bias
{
    extern __shared__ float hwLds[];          // [2][32][256] = 64 KB

    const int tid   = threadIdx.x;
    const int lane  = tid & 31;
    const int wave  = tid >> 5;               // 0..31
    const int m0    = (wave >> 4) * 16;       // 0 or 16
    const int n0    = (wave & 15) * 16;       // 0..240
    const int rc    = lane & 15;              // A-row (m) / B-col (n) within tile
    const int khalf = (lane >> 4) * 2;        // K sub-offset: 0 or 2
    const int vbase = (lane >> 4) * 8;        // D-row base for the 8 acc VGPRs

    // ---------------- layer 1 (both paths) -> LDS, with bias + ReLU --------
    for (int path = 0; path < 2; ++path) {
        const float* __restrict__ W  = path ? bw1 : w1;
        const float* __restrict__ BB = path ? bb1 : b1;
        v8f acc = {};
        for (int k = 0; k < C_DIM; k += 4) {
            // A[m][k'] = latent[(m0+rc)*C + k + khalf + j]
            v2f a = *(const v2f*)(latent + (m0 + rc) * C_DIM + k + khalf);
            // B[k'][n] = W[(n0+rc)*C + k + khalf + j]   (W consumed transposed)
            v2f b = *(const v2f*)(W + (n0 + rc) * C_DIM + k + khalf);
            acc = __builtin_amdgcn_wmma_f32_16x16x4_f32(
                false, a, false, b, (short)0, acc, false, false);
        }
        const float bv = BB[n0 + rc];
        float* __restrict__ dst = hwLds + path * (B_DIM * C_DIM);
#pragma unroll
        for (int v = 0; v < 8; ++v) {
            const int m  = m0 + vbase + v;
            float val    = acc[v] + bv;
            dst[m * C_DIM + (n0 + rc)] = val > 0.0f ? val : 0.0f;
        }
    }

    __syncthreads();

    // ---------------- layer 2 (both paths): LDS -> global workspace --------
    for (int path = 0; path < 2; ++path) {
        const float* __restrict__ W  = path ? bw2 : w2;
        const float* __restrict__ BB = path ? bb2 : b2;
        const float* __restrict__ hw = hwLds + path * (B_DIM * C_DIM);
        v8f acc = {};
        for (int k = 0; k < C_DIM; k += 4) {
            v2f a = *(const v2f*)(hw + (m0 + rc) * C_DIM + k + khalf);
            v2f b = *(const v2f*)(W + (n0 + rc) * C_DIM + k + khalf);
            acc = __builtin_amdgcn_wmma_f32_16x16x4_f32(
                false, a, false, b, (short)0, acc, false, false);
        }
        const float bv = BB[n0 + rc];
        float* __restrict__ dst = wsOut + path * (B_DIM * C_DIM);
#pragma unroll
        for (int v = 0; v < 8; ++v) {
            const int m = m0 + vbase + v;
            dst[m * C_DIM + (n0 + rc)] = acc[v] + bv;
        }
    }
}

// ---------------------------------------------------------------------------
// Streaming apply: out[b,c,:,:] = x[b,c,:,:] * scale[b,c] + bias[b,c].
// 256 MiB of traffic -> pure HBM-bound (~11.5 us at 23.3 TB/s).
// One block per (b,c) plane: 4096 floats = 1024 float4; 256 threads x 4.
// Nontemporal B128 loads/stores (stream >> L2, avoid pollution).
// ---------------------------------------------------------------------------
__global__ __launch_bounds__(256) void adaconv_apply_kernel(
    const float* __restrict__ x,
    const float* __restrict__ wb,   // [2][32][256] from MLP kernel
    float* __restrict__ out)
{
    const int   bc   = blockIdx.x;                 // 0 .. 8191  (b*256 + c)
    const float w    = wb[bc];
    const float b    = wb[B_DIM * C_DIM + bc];
    const size_t base = (size_t)bc * HW_DIM;

    const v4f* __restrict__ x4 = (const v4f*)(x + base);
    v4f*       __restrict__ o4 = (v4f*)(out + base);

#pragma unroll
    for (int i = 0; i < 4; ++i) {
        const int idx = threadIdx.x + i * 256;     // 0..1023 float4 slots
        v4f xv = __builtin_nontemporal_load(x4 + idx);
        v4f ov = xv * w + b;                       // scalar broadcast FMA
        __builtin_nontemporal_store(ov, o4 + idx);
    }
}

extern "C" void kernel_launch(void* const* d_in, const int* in_sizes, int n_in,
                              void* d_out, int out_size, void* d_ws, size_t ws_size,
                              hipStream_t stream) {
    const float* x      = (const float*)d_in[0];
    const float* latent = (const float*)d_in[1];
    const float* w1     = (const float*)d_in[2];
    const float* b1     = (const float*)d_in[3];
    const float* w2     = (const float*)d_in[4];
    const float* b2     = (const float*)d_in[5];
    const float* bw1    = (const float*)d_in[6];
    const float* bb1    = (const float*)d_in[7];
    const float* bw2    = (const float*)d_in[8];
    const float* bb2    = (const float*)d_in[9];

    float* out = (float*)d_out;
    float* ws  = (float*)d_ws;   // 2*32*256 floats = 64 KB of scale/bias

    const size_t ldsBytes = 2 * B_DIM * C_DIM * sizeof(float);  // 64 KB
    adaconv_mlp_kernel<<<1, 1024, ldsBytes, stream>>>(
        latent, w1, b1, w2, b2, bw1, bb1, bw2, bb2, ws);

    adaconv_apply_kernel<<<B_DIM * C_DIM, 256, 0, stream>>>(x, ws, out);
}